// GNN_28329604285033
// MI455X (gfx1250) — compile-verified
//
#include <hip/hip_runtime.h>

typedef __attribute__((ext_vector_type(2))) float v2f;
typedef __attribute__((ext_vector_type(8))) float v8f;

static inline int div_up(int a, int b) { return (a + b - 1) / b; }

// ---------------- zero (float4 grid-stride) ----------------
__global__ __launch_bounds__(256) void k_zero4(float4* __restrict__ p, int n4) {
    int i = blockIdx.x * blockDim.x + threadIdx.x;
    int stride = gridDim.x * blockDim.x;
    for (; i < n4; i += stride) p[i] = make_float4(0.f, 0.f, 0.f, 0.f);
}

// ---------------- degree count ----------------
__global__ __launch_bounds__(256) void k_degree(const int* __restrict__ dst,
                                                float* __restrict__ deg, int E) {
    int i = blockIdx.x * blockDim.x + threadIdx.x;
    int stride = gridDim.x * blockDim.x;
    for (; i < E; i += stride) atomicAdd(&deg[dst[i]], 1.0f);
}

__global__ __launch_bounds__(256) void k_invdeg(float* __restrict__ deg, int n) {
    int i = blockIdx.x * blockDim.x + threadIdx.x;
    int stride = gridDim.x * blockDim.x;
    for (; i < n; i += stride) {
        float d = deg[i];
        deg[i] = 1.0f / fmaxf(d, 1.0f);
    }
}

// ---------------- edge scatter: agg[dst] += x[src], float4 granularity ----------------
template <int DIN4>
__global__ __launch_bounds__(256) void k_scatter4(const float* __restrict__ x,
                                                  const int* __restrict__ src,
                                                  const int* __restrict__ dst,
                                                  float* __restrict__ agg, int E) {
    unsigned total = (unsigned)E * (unsigned)DIN4;
    unsigned i = blockIdx.x * blockDim.x + threadIdx.x;
    unsigned stride = gridDim.x * blockDim.x;
    for (; i < total; i += stride) {
        unsigned e = i / (unsigned)DIN4;        // constant divisor -> mulhi
        unsigned f = i - e * (unsigned)DIN4;
        int s = src[e], d = dst[e];
        const float4 v = *(const float4*)(x + (size_t)s * (DIN4 * 4) + (size_t)f * 4);
        float* p = agg + (size_t)d * (DIN4 * 4) + (size_t)f * 4;
        atomicAdd(p + 0, v.x);
        atomicAdd(p + 1, v.y);
        atomicAdd(p + 2, v.z);
        atomicAdd(p + 3, v.w);
    }
}

// ---------------- fused SAGE layer GEMM via V_WMMA_F32_16X16X4_F32 ----------------
// One wave computes a 64x16 output panel (4 M-subtiles sharing one N-tile):
//   out = (invdeg .* agg) @ Wl^T + bl + xin @ Wr^T  (+ optional ReLU)
// Per K-step: issue all 10 fragment loads (2 weight, 8 activation) as a batch,
// then run 8 WMMAs -> compiler can clause the loads and drain loadcnt
// incrementally while WMMAs execute. All control flow around WMMA is
// wave-uniform -> EXEC stays all-1s as the ISA requires.
template <int DIN>
__global__ __launch_bounds__(256) void k_sage_gemm(
    const float* __restrict__ agg, const float* __restrict__ invdeg,
    const float* __restrict__ xin, const float* __restrict__ Wl,
    const float* __restrict__ bl,  const float* __restrict__ Wr,
    float* __restrict__ out, int n, int dout, int relu)
{
    constexpr int MT = 4;                        // M-subtiles per wave
    const int lane = threadIdx.x & 31;
    const int wave = threadIdx.x >> 5;
    const int ntn  = dout >> 4;                  // N tiles
    const int ntm  = n >> 4;                     // M tiles
    const int mgroups = (ntm + MT - 1) / MT;
    const int total   = mgroups * ntn;
    int tile = blockIdx.x * (blockDim.x >> 5) + wave;
    if (tile >= total) return;                   // whole-wave exit

    const int mg = tile / ntn;
    const int nt = tile - mg * ntn;
    const int n0 = nt << 4;
    const int l15 = lane & 15;
    const int lh  = lane >> 4;                   // f32 A/B layout: lanes 16-31 hold K+2..3
    const int wrow = n0 + l15;                   // output feature = row of W

    const float* lP = Wl + (size_t)wrow * DIN;
    const float* rP = Wr + (size_t)wrow * DIN;

    const float* aP[MT];
    const float* xP[MT];
    float s[MT];
    bool valid[MT];                              // wave-uniform per subtile
#pragma unroll
    for (int t = 0; t < MT; ++t) {
        int mt = mg * MT + t;
        valid[t] = (mt < ntm);
        int m0 = (valid[t] ? mt : (ntm - 1)) << 4;  // clamp: harmless reads, store skipped
        int arow = m0 + l15;
        s[t]  = invdeg[arow];
        aP[t] = agg + (size_t)arow * DIN;
        xP[t] = xin + (size_t)arow * DIN;
    }

    const v8f zero = {0.f, 0.f, 0.f, 0.f, 0.f, 0.f, 0.f, 0.f};
    v8f c[MT];
#pragma unroll
    for (int t = 0; t < MT; ++t) c[t] = zero;

#pragma unroll 2
    for (int k = lh * 2; k < DIN; k += 4) {
        // ---- batch all fragment loads for this K-step ----
        v2f bL = *(const v2f*)(lP + k);          // shared across 4 subtiles
        v2f bR = *(const v2f*)(rP + k);
        v2f aL[MT], aR[MT];
#pragma unroll
        for (int t = 0; t < MT; ++t) {
            aL[t] = *(const v2f*)(aP[t] + k);
            aR[t] = *(const v2f*)(xP[t] + k);
        }
        // ---- then the 8 WMMAs ----
#pragma unroll
        for (int t = 0; t < MT; ++t) {
            v2f a = aL[t];
            a.x *= s[t]; a.y *= s[t];            // fused mean normalization
            c[t] = __builtin_amdgcn_wmma_f32_16x16x4_f32(false, a,     false, bL,
                                                         (short)0, c[t], false, false);
            c[t] = __builtin_amdgcn_wmma_f32_16x16x4_f32(false, aR[t], false, bR,
                                                         (short)0, c[t], false, false);
        }
    }

    const float bias = bl[wrow];
#pragma unroll
    for (int t = 0; t < MT; ++t) {
        if (!valid[t]) continue;                 // wave-uniform
        int m0 = (mg * MT + t) << 4;
        int rbase = m0 + lh * 8;                 // C/D layout: lanes 16-31 hold M=8..15
        float* oP = out + (size_t)rbase * dout + n0 + l15;
#pragma unroll
        for (int r = 0; r < 8; ++r) {
            float v = c[t][r] + bias;
            if (relu) v = fmaxf(v, 0.f);
            oP[(size_t)r * dout] = v;
        }
    }
}

extern "C" void kernel_launch(void* const* d_in, const int* in_sizes, int n_in,
                              void* d_out, int out_size, void* d_ws, size_t ws_size,
                              hipStream_t stream) {
    const float* x   = (const float*)d_in[0];
    const int*   ei  = (const int*)d_in[1];
    const float* Wl1 = (const float*)d_in[2];
    const float* bl1 = (const float*)d_in[3];
    const float* Wr1 = (const float*)d_in[4];
    const float* Wl2 = (const float*)d_in[5];
    const float* bl2 = (const float*)d_in[6];
    const float* Wr2 = (const float*)d_in[7];
    const float* Wl3 = (const float*)d_in[8];
    const float* bl3 = (const float*)d_in[9];
    const float* Wr3 = (const float*)d_in[10];
    float* out = (float*)d_out;

    const int n = in_sizes[0] / 96;   // 50000 (divisible by 16)
    const int E = in_sizes[1] / 2;    // 800000
    const int* src = ei;
    const int* dst = ei + E;

    // workspace layout (floats): invdeg[n] | agg[n*256] | h1[n*256] | h2[n*128]
    float* ws_f   = (float*)d_ws;
    float* invdeg = ws_f;
    float* agg    = invdeg + n;
    float* h1     = agg + (size_t)n * 256;
    float* h2     = h1  + (size_t)n * 256;

    const int TB = 256;
    const int WPB = TB / 32;          // waves per block
    const int MT = 4;

    // --- degrees (shared by all layers) ---
    k_zero4<<<div_up(n / 4, TB), TB, 0, stream>>>((float4*)invdeg, n / 4);
    k_degree<<<div_up(E, TB), TB, 0, stream>>>(dst, invdeg, E);
    k_invdeg<<<div_up(n, TB), TB, 0, stream>>>(invdeg, n);

    // --- layer 1: din=96, dout=256, relu ---
    {
        const int din = 96, dout = 256;
        k_zero4<<<div_up(n * din / 4, TB), TB, 0, stream>>>((float4*)agg, n * din / 4);
        k_scatter4<din / 4><<<4096, TB, 0, stream>>>(x, src, dst, agg, E);
        int tiles = div_up(n / 16, MT) * (dout / 16);
        k_sage_gemm<din><<<div_up(tiles, WPB), TB, 0, stream>>>(
            agg, invdeg, x, Wl1, bl1, Wr1, h1, n, dout, 1);
    }
    // --- layer 2: din=256, dout=128, relu ---
    {
        const int din = 256, dout = 128;
        k_zero4<<<div_up(n * din / 4, TB), TB, 0, stream>>>((float4*)agg, n * din / 4);
        k_scatter4<din / 4><<<4096, TB, 0, stream>>>(h1, src, dst, agg, E);
        int tiles = div_up(n / 16, MT) * (dout / 16);
        k_sage_gemm<din><<<div_up(tiles, WPB), TB, 0, stream>>>(
            agg, invdeg, h1, Wl2, bl2, Wr2, h2, n, dout, 1);
    }
    // --- layer 3: din=128, dout=64, no relu ---
    {
        const int din = 128, dout = 64;
        k_zero4<<<div_up(n * din / 4, TB), TB, 0, stream>>>((float4*)agg, n * din / 4);
        k_scatter4<din / 4><<<4096, TB, 0, stream>>>(h2, src, dst, agg, E);
        int tiles = div_up(n / 16, MT) * (dout / 16);
        k_sage_gemm<din><<<div_up(tiles, WPB), TB, 0, stream>>>(
            agg, invdeg, h2, Wl3, bl3, Wr3, out, n, dout, 0);
    }
}